// Predictor_65429531787931
// MI455X (gfx1250) — compile-verified
//
#include <hip/hip_runtime.h>

typedef float v2f __attribute__((ext_vector_type(2)));
typedef float v8f __attribute__((ext_vector_type(8)));

#define IN_DIM 128
#define NCLS   2

// ---------------------------------------------------------------------------
// Kernel 1: per-node projection  proj[n][0:4] = [x.Wsrc0, x.Wsrc1, x.Wdst0, x.Wdst1]
// GEMM [n_nodes x 128] * [128 x 4(pad16)] via V_WMMA_F32_16X16X4_F32.
// One 16-node tile per wave32; 8 waves per 256-thread block.
// ---------------------------------------------------------------------------
__global__ __launch_bounds__(256) void node_proj_kernel(
    const float* __restrict__ x, const float* __restrict__ W,
    float* __restrict__ proj, int n_nodes, int n_tiles) {
  // Stage B matrix (K=128 x N=16, cols 0..3 real, rest zero) in LDS: 8 KB.
  __shared__ float ldsB[IN_DIM * 16];
  const int tid = threadIdx.x;
  for (int idx = tid; idx < IN_DIM * 16; idx += 256) {
    const int k = idx >> 4;
    const int n = idx & 15;
    float v = 0.0f;
    if (n < NCLS)            v = W[n * (2 * IN_DIM) + k];            // src half
    else if (n < 2 * NCLS)   v = W[(n - NCLS) * (2 * IN_DIM) + IN_DIM + k]; // dst half
    ldsB[idx] = v;
  }
  __syncthreads();

  const int wave = tid >> 5;
  const int lane = tid & 31;
  const int tile = blockIdx.x * 8 + wave;
  if (tile >= n_tiles) return;              // wave-uniform: EXEC stays all-1s

  const int half = lane >> 4;               // 0 -> K{k,k+1}, 1 -> K{k+2,k+3}
  const int mn   = lane & 15;               // A: row M ; B: col N
  int row = tile * 16 + mn;
  if (row >= n_nodes) row = n_nodes - 1;    // clamp partial tile (re-read, unused)
  const float* __restrict__ xrow = x + (size_t)row * IN_DIM;

  v8f acc = {};
#pragma unroll
  for (int k = 0; k < IN_DIM; k += 4) {
    const int kk = k + 2 * half;
    v2f a, bb;
    a.x  = xrow[kk];
    a.y  = xrow[kk + 1];
    bb.x = ldsB[kk * 16 + mn];
    bb.y = ldsB[(kk + 1) * 16 + mn];
    // D = A(16x4 f32) * B(4x16 f32) + C   -- full f32 precision
    acc = __builtin_amdgcn_wmma_f32_16x16x4_f32(
        /*neg_a=*/false, a, /*neg_b=*/false, bb,
        /*c_mod=*/(short)0, acc, /*reuse_a=*/false, /*reuse_b=*/false);
  }

  // C/D layout: VGPR r -> M = r + 8*half ; N = mn (lanes 0..15 / 16..31).
  if (mn < 2 * NCLS) {
#pragma unroll
    for (int r = 0; r < 8; ++r) {
      const int M = tile * 16 + r + 8 * half;
      if (M < n_nodes) proj[(size_t)M * 4 + mn] = acc[r];
    }
  }
}

// ---------------------------------------------------------------------------
// Kernel 2: per-edge combine. Two 16B gathers from the 1.6 MB proj table
// (L2-resident on MI455X: 192 MB L2), coalesced float2 store.
// ---------------------------------------------------------------------------
__global__ __launch_bounds__(256) void edge_score_kernel(
    const float4* __restrict__ proj, const int* __restrict__ ei,
    const float* __restrict__ bias, float2* __restrict__ out, int E) {
  const int e = blockIdx.x * 256 + threadIdx.x;
  if (e >= E) return;
  const int s = ei[e];       // edge_index row 0 (src)
  const int d = ei[E + e];   // edge_index row 1 (dst)
  const float4 p = proj[s];
  const float4 q = proj[d];
  float2 r;
  r.x = p.x + q.z + bias[0];
  r.y = p.y + q.w + bias[1];
  out[e] = r;
}

// ---------------------------------------------------------------------------
extern "C" void kernel_launch(void* const* d_in, const int* in_sizes, int n_in,
                              void* d_out, int out_size, void* d_ws, size_t ws_size,
                              hipStream_t stream) {
  const float* x    = (const float*)d_in[0];   // [n_nodes, 128] f32
  const int*   ei   = (const int*)d_in[1];     // [2, E] int32
  const float* W    = (const float*)d_in[2];   // [2, 256] f32
  const float* bias = (const float*)d_in[3];   // [2] f32

  const int n_nodes = in_sizes[0] / IN_DIM;
  const int E       = in_sizes[1] / 2;

  float* proj = (float*)d_ws;                  // n_nodes * 4 f32 = 1.6 MB scratch

  const int n_tiles = (n_nodes + 15) / 16;
  const int grid1   = (n_tiles + 7) / 8;       // 8 wave32 tiles per block
  node_proj_kernel<<<grid1, 256, 0, stream>>>(x, W, proj, n_nodes, n_tiles);

  const int grid2 = (E + 255) / 256;
  edge_score_kernel<<<grid2, 256, 0, stream>>>((const float4*)proj, ei, bias,
                                               (float2*)d_out, E);
}